// LIFLayer_9972914061671
// MI455X (gfx1250) — compile-verified
//
#include <hip/hip_runtime.h>

// LIF spiking layer: out[t,b,f] = spike, serial recurrence over t.
//   v = 0.9*v + x[t];  s = (v >= 1.0) ? 1.0 : 0.0;  v -= s;  out[t] = s
// Shapes: x, out are [50, 128, 4096] float32 (time-major).
//
// Memory-bound stream: 105 MB read + 105 MB written. HBM floor ~9 us @ 23.3 TB/s.
// x (105 MB) fits in the 192 MB L2 -> loads use default temporal policy so x
// stays L2-resident across graph replays; stores use NT so the output stream
// does not evict it. 8-deep per-wave load ring keeps ~16 MB of B128 loads in
// flight device-wide (Little's law target for 23.3 TB/s at ~700 ns latency).

typedef float v4f __attribute__((ext_vector_type(4)));

#define NB_STEPS   50
#define BATCH      128
#define FEATURES   4096
#define N_PER_STEP (BATCH * FEATURES)      // 524288 floats per timestep
#define VEC_STRIDE (N_PER_STEP / 4)        // 131072 float4 per timestep
#define DECAY      0.9f
#define THRESH     1.0f
#define DEPTH      8                        // in-flight load ring depth
#define PF_DIST    16                       // L2 prefetch distance (timesteps)

// One LIF step on a float4 lane-bundle. Vector ops for the arithmetic,
// scalar ternaries (rvalue element reads) for the Heaviside select.
__device__ __forceinline__ v4f lif_step4(v4f& v, v4f cur) {
    v = DECAY * v + cur;                    // leak + integrate (v_pk_fma_f32)
    v4f s;
    s.x = (v.x >= THRESH) ? THRESH : 0.0f;
    s.y = (v.y >= THRESH) ? THRESH : 0.0f;
    s.z = (v.z >= THRESH) ? THRESH : 0.0f;
    s.w = (v.w >= THRESH) ? THRESH : 0.0f;
    v = v - s;                              // soft reset
    return s;                               // spike * threshold (threshold = 1)
}

__global__ __launch_bounds__(256, 1)
void lif_stream_kernel(const float* __restrict__ xf, float* __restrict__ outf)
{
    const int idx = blockIdx.x * blockDim.x + threadIdx.x;   // 0 .. VEC_STRIDE-1
    const v4f* __restrict__ x   = (const v4f*)xf;
    v4f*       __restrict__ out = (v4f*)outf;

    // Prologue: fill the 8-deep ring of in-flight B128 loads (temporal policy:
    // let x become / stay L2-resident).
    v4f buf[DEPTH];
    #pragma unroll
    for (int i = 0; i < DEPTH; ++i)
        buf[i] = x[idx + i * VEC_STRIDE];

    v4f v = (v4f)0.0f;

    // Fully unrolled: every buf index is a compile-time constant -> the ring
    // lives entirely in VGPRs and each iteration retires one load and issues
    // the next, keeping DEPTH loads outstanding per wave.
    #pragma unroll
    for (int t = 0; t < NB_STEPS; ++t) {
        if (t + PF_DIST < NB_STEPS) {
            // global_prefetch_b8: warm L2 ahead of the load stream (1st pass).
            __builtin_prefetch(&x[idx + (t + PF_DIST) * VEC_STRIDE], 0, 0);
        }

        v4f cur = buf[t % DEPTH];
        v4f s = lif_step4(v, cur);
        // NT store: stream the output past L2 so x stays resident.
        __builtin_nontemporal_store(s, &out[idx + t * VEC_STRIDE]);

        if (t + DEPTH < NB_STEPS)
            buf[t % DEPTH] = x[idx + (t + DEPTH) * VEC_STRIDE];
    }
}

extern "C" void kernel_launch(void* const* d_in, const int* in_sizes, int n_in,
                              void* d_out, int out_size, void* d_ws, size_t ws_size,
                              hipStream_t stream) {
    (void)in_sizes; (void)n_in; (void)out_size; (void)d_ws; (void)ws_size;

    const float* x   = (const float*)d_in[0];   // [50, 128, 4096] f32
    float*       out = (float*)d_out;           // [50, 128, 4096] f32

    const int threads = 256;                    // 8 wave32s per block
    const int blocks  = VEC_STRIDE / threads;   // 131072 / 256 = 512 blocks

    lif_stream_kernel<<<blocks, threads, 0, stream>>>(x, out);
}